// OutputLayer_19267223290690
// MI455X (gfx1250) — compile-verified
//
#include <hip/hip_runtime.h>
#include <math.h>

// ---------------------------------------------------------------------------
// Outputs (flat in d_out):   [0, n_bags)            max_weights
//                            [n_bags, n_bags+n_src) per_src
// Workspace (d_ws):          lin[n_src] floats, bag_max[n_bags] floats
// ---------------------------------------------------------------------------

typedef __attribute__((ext_vector_type(2))) float v2f;
typedef __attribute__((ext_vector_type(8))) float v8f;

#define NEG_INF (-__builtin_inff())

// fp32 atomic max via monotone int mapping:
//   sign bit clear -> signed atomicMax ; sign bit set -> unsigned atomicMin.
// Correct for all float orderings (incl. -0/+0); accumulator is init'd to -inf
// and only ever holds values written through this function.
__device__ __forceinline__ void atomic_fmax(float* addr, float v) {
    const int iv = __float_as_int(v);
    if (iv >= 0) atomicMax((int*)addr, iv);
    else         atomicMin((unsigned int*)addr, (unsigned int)iv);
}

// One edge: v = weights[dst] * lin[src]; segment-max into per_src[src].
// Plain-load pre-filter: per_src is monotone non-decreasing, so a stale read
// can only cause an extra (harmless) atomic, never a missed update.
__device__ __forceinline__ void edge_update(int s, int d,
                                            const float* __restrict__ weights,
                                            const float* __restrict__ lin,
                                            float* __restrict__ per_src) {
    const float v = weights[d] * lin[s];
    if (v > per_src[s]) atomic_fmax(&per_src[s], v);
}

// Kernel 1: lin = feat @ W_lin^T via V_WMMA_F32_16X16X4_F32 (16 rows / wave),
// plus -inf init of both segment-max accumulators (grid covers both sizes).
__global__ void lin_wmma_init_kernel(const float* __restrict__ feat,
                                     const float* __restrict__ W_lin,
                                     float* __restrict__ lin,
                                     float* __restrict__ per_src,
                                     float* __restrict__ bag_max,
                                     int n_src, int n_bags)
{
    const int tid  = blockIdx.x * blockDim.x + threadIdx.x;
    const int lane = threadIdx.x & 31;            // wave32
    const int base = (tid >> 5) * 16;             // 16 rows per wave
    const int row  = base + (lane & 15);
    const bool lo  = lane < 16;

    // A (16x4 f32, MxK): lanes 0-15 carry {K=0 in v0, K=1 in v1} for M=lane,
    // lanes 16-31 carry {K=2, K=3} -> zero-pad (FEAT_DIM == 2).
    const bool ld = lo && (row < n_src);
    v2f a;
    a.x = ld ? feat[2 * row + 0] : 0.0f;
    a.y = ld ? feat[2 * row + 1] : 0.0f;

    // B (4x16 f32, KxN): every column n set to W_lin, so D[m][n] == lin[m].
    // Lanes 0-15 hold rows K=0 (v0) / K=1 (v1); lanes 16-31 (K=2,3) -> zero.
    const float w0 = W_lin[0];
    const float w1 = W_lin[1];
    v2f b;
    b.x = lo ? w0 : 0.0f;
    b.y = lo ? w1 : 0.0f;

    v8f c = {};
    // No early returns above -> EXEC is all 1s at the WMMA (required).
    c = __builtin_amdgcn_wmma_f32_16x16x4_f32(
            /*neg_a=*/false, a, /*neg_b=*/false, b,
            /*c_mod=*/(short)0, c, /*reuse_a=*/false, /*reuse_b=*/false);

    // D (16x16 f32): VGPR r holds row (base+r) on lanes 0-15 and row
    // (base+8+r) on lanes 16-31; all 16 columns identical -> spill from
    // lanes 0 and 16 only.
    if (lane == 0) {
#pragma unroll
        for (int r = 0; r < 8; ++r)
            if (base + r < n_src) lin[base + r] = c[r];
    } else if (lane == 16) {
#pragma unroll
        for (int r = 0; r < 8; ++r)
            if (base + 8 + r < n_src) lin[base + 8 + r] = c[r];
    }

    // Init segment-max accumulators to -inf (matches jax segment_max fill).
    // Launch grid guarantees gridDim.x*blockDim.x >= max(n_src, n_bags).
    if (tid < n_src)  per_src[tid] = NEG_INF;
    if (tid < n_bags) bag_max[tid] = NEG_INF;
}

// Kernel 2 (hot path): 4 edges per thread via int4 index loads (16B coalesced
// transactions on the two streamed arrays); weights/lin gathers stay
// L2-resident (400 KB each). Tail edges handled by the first few threads.
__global__ void edge_kernel(const int*   __restrict__ edge_src,
                            const int*   __restrict__ edge_dst,
                            const float* __restrict__ weights,
                            const float* __restrict__ lin,
                            float*       __restrict__ per_src,
                            int n_edges)
{
    const int i  = blockIdx.x * blockDim.x + threadIdx.x;
    const int n4 = n_edges >> 2;

    if (i < n4) {
        const int4 s4 = ((const int4*)edge_src)[i];
        const int4 d4 = ((const int4*)edge_dst)[i];
        edge_update(s4.x, d4.x, weights, lin, per_src);
        edge_update(s4.y, d4.y, weights, lin, per_src);
        edge_update(s4.z, d4.z, weights, lin, per_src);
        edge_update(s4.w, d4.w, weights, lin, per_src);
    }
    // tail (n_edges % 4 edges), handled by threads 0..2 of the grid
    const int tail = n_edges & 3;
    if (i < tail) {
        const int e = n4 * 4 + i;
        edge_update(edge_src[e], edge_dst[e], weights, lin, per_src);
    }
}

// Kernel 3: per-bag segment max over sources (~10 sources/bag).
__global__ void bag_kernel(const float* __restrict__ per_src,
                           const int*   __restrict__ bag_of_source,
                           float*       __restrict__ bag_max,
                           int n_src)
{
    const int i = blockIdx.x * blockDim.x + threadIdx.x;
    if (i >= n_src) return;
    const float v = per_src[i];
    const int   b = bag_of_source[i];
    if (v > bag_max[b]) atomic_fmax(&bag_max[b], v);
}

// Kernel 4: sentinel mapping (empty bag -> -inf -> 0), matches torch init.
__global__ void finalize_kernel(const float* __restrict__ bag_max,
                                float*       __restrict__ out,
                                int n_bags)
{
    const int i = blockIdx.x * blockDim.x + threadIdx.x;
    if (i >= n_bags) return;
    const float m = bag_max[i];
    out[i] = (m > -10.0f) ? m : 0.0f;
}

extern "C" void kernel_launch(void* const* d_in, const int* in_sizes, int n_in,
                              void* d_out, int out_size, void* d_ws, size_t ws_size,
                              hipStream_t stream)
{
    const float* weights       = (const float*)d_in[0];   // [n_w, 1]
    const float* feat          = (const float*)d_in[1];   // [n_src, 2]
    const float* W_lin         = (const float*)d_in[2];   // [1, 2]
    const int*   edge_src      = (const int*)  d_in[3];   // [n_edges]
    const int*   edge_dst      = (const int*)  d_in[4];   // [n_edges]
    const int*   bag_of_source = (const int*)  d_in[5];   // [n_src]
    (void)n_in; (void)ws_size;

    const int n_src   = in_sizes[5];
    const int n_edges = in_sizes[3];
    const int n_bags  = out_size - n_src;   // num_bags is a device scalar; derive on host

    float* out_bags = (float*)d_out;            // [n_bags]
    float* per_src  = (float*)d_out + n_bags;   // [n_src]
    float* lin      = (float*)d_ws;             // [n_src]
    float* bag_max  = (float*)d_ws + n_src;     // [n_bags]

    const int T = 256;  // 8 wave32s per block
    // WMMA part needs ceil(n_src/128) blocks (16 rows/wave * 8 waves);
    // init part needs coverage of max(n_src, n_bags) threads.
    int grid_lin = (n_src + 127) / 128;
    const int grid_init = ((n_src > n_bags ? n_src : n_bags) + T - 1) / T;
    if (grid_init > grid_lin) grid_lin = grid_init;

    const int n4 = n_edges >> 2;
    int grid_edge = (n4 + T - 1) / T;
    if (grid_edge < 1) grid_edge = 1;

    const int grid_src = (n_src + T - 1) / T;
    const int grid_bag = (n_bags + T - 1) / T;

    lin_wmma_init_kernel<<<grid_lin, T, 0, stream>>>(feat, W_lin, lin, per_src,
                                                     bag_max, n_src, n_bags);
    edge_kernel<<<grid_edge, T, 0, stream>>>(edge_src, edge_dst, weights, lin,
                                             per_src, n_edges);
    bag_kernel<<<grid_src, T, 0, stream>>>(per_src, bag_of_source, bag_max, n_src);
    finalize_kernel<<<grid_bag, T, 0, stream>>>(bag_max, out_bags, n_bags);
}